// nKAN_7722351198983
// MI455X (gfx1250) — compile-verified
//
#include <hip/hip_runtime.h>

typedef float v2f __attribute__((ext_vector_type(2)));
typedef float v8f __attribute__((ext_vector_type(8)));

#define IDIM 512
#define ODIM 512
#define HDIM 16
#define WSTRIDE 288   // 16*18 floats per (i,o) edge block (contiguous)

__global__ void nkan_init_out(const float* __restrict__ bias2, float* __restrict__ out) {
    int o = blockIdx.x * blockDim.x + threadIdx.x;
    if (o < ODIM) out[o] = bias2[o];
}

// One wave32 per (o, i-chunk). Each wave streams 32 edge blocks of 1152B.
// Per edge: h1 = relu(x*W1+b0); h2 = relu(h1 @ W2 + b1) via 4x V_WMMA_F32_16X16X4_F32
// (A = W2^T chunk, B = h1 in column 0); partial = sum_h h2*W3 accumulated, one atomic/wave.
__global__ __launch_bounds__(256) void nkan_kernel(
    const float* __restrict__ x,
    const float* __restrict__ weight,
    const float* __restrict__ bias0,
    const float* __restrict__ bias1,
    float* __restrict__ out)
{
    const int tid  = blockIdx.x * blockDim.x + threadIdx.x;
    const int lane = threadIdx.x & 31;
    const int gid  = tid >> 5;             // global wave id: 0..8191
    const int o    = gid & (ODIM - 1);     // adjacent waves -> adjacent o blocks (contiguous)
    const int ic   = gid >> 9;             // 0..15, chunk of 32 i's
    const int m    = lane & 15;            // A-matrix M row / D column N
    const int half = lane >> 4;            // K sub-block selector per ISA A/B layout
    const bool col0 = (m == 0);            // lanes 0 and 16 carry B column N=0

    float acc = 0.0f;

    for (int ii = 0; ii < 32; ++ii) {
        const int i = ic * 32 + ii;
        const float* __restrict__ W = weight + (size_t)(i * ODIM + o) * WSTRIDE;
        // prefetch next i's block for this o (gfx1250 global_prefetch_b8)
        __builtin_prefetch(W + (size_t)ODIM * WSTRIDE, 0, 0);

        const float xi  = x[i];
        const float b0  = bias0[i * ODIM + o];
        const float b1v = bias1[i * ODIM + o];

        // Gather A (W2 transposed into ISA 16x4 layout) and compute h1 values
        // this lane would supply for the B operand.
        float av[4][2];
        float hv[4][2];
        #pragma unroll
        for (int cc = 0; cc < 4; ++cc) {
            #pragma unroll
            for (int v = 0; v < 2; ++v) {
                const int k = cc * 4 + 2 * half + v;      // contraction (hidden) index
                av[cc][v] = W[k * 18 + 1 + m];            // A[m,kk] = W2[k,m]
                const float h = fmaf(xi, W[k * 18], b0);  // h1[k] pre-relu
                hv[cc][v] = h > 0.0f ? h : 0.0f;
            }
        }

        // h2 = W2^T * h1 via 4 chained 16x16x4 f32 WMMAs; result in column N=0.
        v8f c = {};
        #pragma unroll
        for (int cc = 0; cc < 4; ++cc) {
            v2f a, b;
            a.x = av[cc][0];
            a.y = av[cc][1];
            b.x = col0 ? hv[cc][0] : 0.0f;   // B[kk, n] = h1[k] iff n==0
            b.y = col0 ? hv[cc][1] : 0.0f;
            c = __builtin_amdgcn_wmma_f32_16x16x4_f32(
                    /*neg_a=*/false, a, /*neg_b=*/false, b,
                    /*c_mod=*/(short)0, c, /*reuse_a=*/false, /*reuse_b=*/false);
        }

        // Stage 3: lanes 0/16 hold D[m,0] for m = r + 8*half in c[r].
        float s = 0.0f;
        #pragma unroll
        for (int r = 0; r < 8; ++r) {
            float h2 = c[r] + b1v;
            h2 = h2 > 0.0f ? h2 : 0.0f;
            s = fmaf(h2, W[(r + 8 * half) * 18 + 17], s);
        }
        acc += col0 ? s : 0.0f;
    }

    // Only lanes 0 and 16 carry valid partials.
    const float total = __shfl(acc, 0, 32) + __shfl(acc, 16, 32);
    if (lane == 0) atomicAdd(&out[o], total);
}

extern "C" void kernel_launch(void* const* d_in, const int* in_sizes, int n_in,
                              void* d_out, int out_size, void* d_ws, size_t ws_size,
                              hipStream_t stream) {
    const float* x      = (const float*)d_in[0];
    const float* weight = (const float*)d_in[1];
    const float* bias0  = (const float*)d_in[2];
    const float* bias1  = (const float*)d_in[3];
    const float* bias2  = (const float*)d_in[4];
    float* out = (float*)d_out;

    (void)in_sizes; (void)n_in; (void)out_size; (void)d_ws; (void)ws_size;

    nkan_init_out<<<2, 256, 0, stream>>>(bias2, out);
    // 8192 waves total: 1024 blocks x 256 threads (8 waves each)
    nkan_kernel<<<1024, 256, 0, stream>>>(x, weight, bias0, bias1, out);
}